// EDSim_33509334843946
// MI455X (gfx1250) — compile-verified
//
#include <hip/hip_runtime.h>

#define N_Q  2048
#define M_K  1024
#define DIN  128
#define DK   64

typedef __attribute__((ext_vector_type(2))) float v2f;
typedef __attribute__((ext_vector_type(8))) float v8f;

// ---------------------------------------------------------------------------
// Kernel 1: row-wise projection  XP[row,:] = X[row,:] @ W^T
// X: [rows, DIN], W: [DK, DIN] row-major, XP: [rows, DK]
// One block of 64 threads per row; q-row staged in LDS, W streams from L2.
// ---------------------------------------------------------------------------
__global__ __launch_bounds__(64) void proj_kernel(const float* __restrict__ X,
                                                  const float* __restrict__ W,
                                                  float* __restrict__ XP) {
    __shared__ float xs[DIN];
    const int row = blockIdx.x;
    const int t   = threadIdx.x;                 // 0..63
    xs[t]      = X[row * DIN + t];
    xs[t + 64] = X[row * DIN + t + 64];
    __syncthreads();
    const float* w = W + t * DIN;
    float acc = 0.f;
#pragma unroll 8
    for (int d = 0; d < DIN; ++d) acc = fmaf(xs[d], w[d], acc);
    XP[row * DK + t] = acc;
}

// online-softmax pair combine: (m,s) <- merge (m,s),(om,os)
__device__ __forceinline__ void sm_combine(float& m, float& s, float om, float os) {
    float nm = fmaxf(m, om);
    s = s * __expf(m - nm) + os * __expf(om - nm);
    m = nm;
}

// ---------------------------------------------------------------------------
// Kernel 2: fused distance GEMM (f32 WMMA) + online softmax over m.
// Each wave owns one 16-row n-tile; 4 waves per block.
// dist²[n,m] = |QP[n]|² + |KP[m]|² - 2·(QP·KPᵀ)[n,m]   (WMMA 16x16x4 f32, K=64)
// Pass 1 accumulates per-row running (max,sum); pass 2 recomputes the tile
// scores and writes exp(score-max)/sum. No LDS, scores never hit memory.
// ---------------------------------------------------------------------------
__global__ __launch_bounds__(128) void dist_softmax_kernel(
    const float* __restrict__ QP, const float* __restrict__ KP,
    float* __restrict__ out) {
    const int lane  = threadIdx.x & 31;
    const int wv    = threadIdx.x >> 5;
    const int ntile = blockIdx.x * 4 + wv;
    const int half  = lane >> 4;     // K-half select for A/B fragments
    const int lr    = lane & 15;     // row (A) / col (B,C) within tile

    // ---- A fragments: QP rows of this n-tile (ISA 16x4 f32 A layout) ----
    const float* qrow = QP + (ntile * 16 + lr) * DK;
    v2f a[16];
    float qn_part = 0.f;
#pragma unroll
    for (int s = 0; s < 16; ++s) {
        a[s] = *(const v2f*)(qrow + 4 * s + 2 * half);
        qn_part = fmaf(a[s].x, a[s].x, qn_part);
        qn_part = fmaf(a[s].y, a[s].y, qn_part);
    }
    // |QP[row]|² for row = ntile*16+lr (both K-halves combined)
    float qn = qn_part + __shfl_xor(qn_part, 16, 32);
    // C-slot v maps to row v + 8*half -> broadcast that row's norm
    float qnv[8];
#pragma unroll
    for (int v = 0; v < 8; ++v) qnv[v] = __shfl(qn, v + 8 * half, 32);

    float mx[8], sm[8];
#pragma unroll
    for (int v = 0; v < 8; ++v) { mx[v] = -3.0e38f; sm[v] = 0.f; }

    // ================= pass 1: online softmax statistics =================
    for (int mt = 0; mt < M_K / 16; ++mt) {
        const float* krow = KP + (mt * 16 + lr) * DK;   // KPᵀ column
        v2f b[16];
        float kn_part = 0.f;
#pragma unroll
        for (int s = 0; s < 16; ++s) {
            b[s] = *(const v2f*)(krow + 4 * s + 2 * half);
            kn_part = fmaf(b[s].x, b[s].x, kn_part);
            kn_part = fmaf(b[s].y, b[s].y, kn_part);
        }
        float kn = kn_part + __shfl_xor(kn_part, 16, 32); // |KP[col]|²

        v8f c = {0.f, 0.f, 0.f, 0.f, 0.f, 0.f, 0.f, 0.f};
#pragma unroll
        for (int s = 0; s < 16; ++s)
            c = __builtin_amdgcn_wmma_f32_16x16x4_f32(false, a[s], false, b[s],
                                                      (short)0, c, false, false);
#pragma unroll
        for (int v = 0; v < 8; ++v) {
            float d2    = fmaxf(qnv[v] + kn - 2.f * c[v], 0.f);
            float score = -0.5f * sqrtf(d2);
            float nm    = fmaxf(mx[v], score);
            sm[v] = sm[v] * __expf(mx[v] - nm) + __expf(score - nm);
            mx[v] = nm;
        }
    }

    // reduce (max,sum) across the 16 lanes that share each row
#pragma unroll
    for (int v = 0; v < 8; ++v) {
#pragma unroll
        for (int mask = 1; mask <= 8; mask <<= 1) {
            float om = __shfl_xor(mx[v], mask, 32);
            float os = __shfl_xor(sm[v], mask, 32);
            sm_combine(mx[v], sm[v], om, os);
        }
    }
    float rcp[8];
#pragma unroll
    for (int v = 0; v < 8; ++v) rcp[v] = 1.0f / sm[v];

    // ================= pass 2: recompute + write normalized ==============
    for (int mt = 0; mt < M_K / 16; ++mt) {
        const float* krow = KP + (mt * 16 + lr) * DK;
        v2f b[16];
        float kn_part = 0.f;
#pragma unroll
        for (int s = 0; s < 16; ++s) {
            b[s] = *(const v2f*)(krow + 4 * s + 2 * half);
            kn_part = fmaf(b[s].x, b[s].x, kn_part);
            kn_part = fmaf(b[s].y, b[s].y, kn_part);
        }
        float kn = kn_part + __shfl_xor(kn_part, 16, 32);

        v8f c = {0.f, 0.f, 0.f, 0.f, 0.f, 0.f, 0.f, 0.f};
#pragma unroll
        for (int s = 0; s < 16; ++s)
            c = __builtin_amdgcn_wmma_f32_16x16x4_f32(false, a[s], false, b[s],
                                                      (short)0, c, false, false);
#pragma unroll
        for (int v = 0; v < 8; ++v) {
            float d2    = fmaxf(qnv[v] + kn - 2.f * c[v], 0.f);
            float score = -0.5f * sqrtf(d2);
            int   row   = ntile * 16 + v + 8 * half;
            out[row * M_K + mt * 16 + lr] = __expf(score - mx[v]) * rcp[v];
        }
    }
}

// ---------------------------------------------------------------------------
extern "C" void kernel_launch(void* const* d_in, const int* in_sizes, int n_in,
                              void* d_out, int out_size, void* d_ws, size_t ws_size,
                              hipStream_t stream) {
    (void)in_sizes; (void)n_in; (void)out_size; (void)ws_size;
    const float* q = (const float*)d_in[0];   // [N_Q, DIN]
    const float* k = (const float*)d_in[1];   // [M_K, DIN]
    const float* W = (const float*)d_in[2];   // [DK,  DIN]
    float* out = (float*)d_out;               // [N_Q, M_K]

    float* QP = (float*)d_ws;                 // [N_Q, DK]  512 KB
    float* KP = QP + (size_t)N_Q * DK;        // [M_K, DK]  256 KB

    proj_kernel<<<N_Q, 64, 0, stream>>>(q, W, QP);
    proj_kernel<<<M_K, 64, 0, stream>>>(k, W, KP);
    dist_softmax_kernel<<<(N_Q / 16) / 4, 128, 0, stream>>>(QP, KP, out);
}